// SWA_33208687133083
// MI455X (gfx1250) — compile-verified
//
#include <hip/hip_runtime.h>
#include <hip/hip_bf16.h>
#include <math.h>

typedef __attribute__((ext_vector_type(2))) float v2f;
typedef __attribute__((ext_vector_type(8))) float v8f;

#define BATCH 16
#define DIMC  512
#define S     128

// ---------------------------------------------------------------------------
// Kernel 1: 1x1 conv 512 -> 2 channels.  Pure bandwidth: 512 MB of x streamed
// once with b128 loads, weights (2x512 floats) staged in LDS.
// grid = 256 blocks * 256 threads; each thread owns 4 consecutive w (float4),
// each 32-thread wave owns one (b,h) row.
// ---------------------------------------------------------------------------
__global__ __launch_bounds__(256) void k1_pointwise(
    const float* __restrict__ x, const float* __restrict__ wc,
    float* __restrict__ x2) {
  __shared__ float sw0[DIMC];
  __shared__ float sw1[DIMC];
  int t = threadIdx.x;
  for (int i = t; i < DIMC; i += 256) { sw0[i] = wc[i]; sw1[i] = wc[DIMC + i]; }
  __syncthreads();

  int r  = t >> 5;          // row within block: 0..7
  int q  = t & 31;          // float4 slot within row: 0..31
  int bh = blockIdx.x * 8 + r;   // 0..2047
  int b  = bh >> 7;
  int h  = bh & 127;

  const float* xp = x + ((b * DIMC) * (S * S)) + h * S + q * 4;
  float a0x = 0.f, a0y = 0.f, a0z = 0.f, a0w = 0.f;
  float a1x = 0.f, a1y = 0.f, a1z = 0.f, a1w = 0.f;
  for (int c = 0; c < DIMC; ++c) {
    float4 v = *(const float4*)(xp + c * (S * S));
    float w0 = sw0[c], w1 = sw1[c];
    a0x = fmaf(v.x, w0, a0x); a0y = fmaf(v.y, w0, a0y);
    a0z = fmaf(v.z, w0, a0z); a0w = fmaf(v.w, w0, a0w);
    a1x = fmaf(v.x, w1, a1x); a1y = fmaf(v.y, w1, a1y);
    a1z = fmaf(v.z, w1, a1z); a1w = fmaf(v.w, w1, a1w);
  }
  float4 o0 = make_float4(a0x, a0y, a0z, a0w);
  float4 o1 = make_float4(a1x, a1y, a1z, a1w);
  int base0 = ((b * 2 + 0) * S + h) * S + q * 4;
  int base1 = ((b * 2 + 1) * S + h) * S + q * 4;
  *(float4*)(x2 + base0) = o0;
  *(float4*)(x2 + base1) = o1;
}

// ---------------------------------------------------------------------------
// Kernel 2: the two squeeze convs.  grid = 16 blocks (one per batch),
// 128 threads.  Weights (2*1280 floats) staged in LDS; x2[b] (128 KB) is
// L2-resident.
//   ww_pre[b,w] = sum_{c,kh,kk} x2[b,c,kh,w+kk-2] * Www[c,kh,kk]   (pad w)
//   hw_pre[b,h] = sum_{c,kk,kw} x2[b,c,h+kk-2,kw] * Whw[c,kk,kw]   (pad h)
// ---------------------------------------------------------------------------
__global__ __launch_bounds__(128) void k2_squeeze(
    const float* __restrict__ x2,
    const float* __restrict__ w_ww,   // [2,128,5]
    const float* __restrict__ w_hw,   // [2,5,128]
    float* __restrict__ ww_pre, float* __restrict__ hw_pre) {
  __shared__ float sWw[2 * S * 5];
  __shared__ float sWh[2 * 5 * S];
  int t = threadIdx.x;
  int b = blockIdx.x;
  for (int i = t; i < 2 * S * 5; i += 128) { sWw[i] = w_ww[i]; sWh[i] = w_hw[i]; }
  __syncthreads();

  // ww branch: thread t = output column w
  float acc = 0.f;
  for (int c = 0; c < 2; ++c) {
    for (int kh = 0; kh < S; ++kh) {
      const float* xrow = x2 + ((b * 2 + c) * S + kh) * S;
      const float* wrow = sWw + (c * S + kh) * 5;
      #pragma unroll
      for (int kk = 0; kk < 5; ++kk) {
        int wi = t + kk - 2;
        if (wi >= 0 && wi < S) acc = fmaf(xrow[wi], wrow[kk], acc);
      }
    }
  }
  ww_pre[b * S + t] = acc;

  // hw branch: thread t = output row h
  float acc2 = 0.f;
  for (int c = 0; c < 2; ++c) {
    #pragma unroll
    for (int kk = 0; kk < 5; ++kk) {
      int hr = t + kk - 2;
      if (hr < 0 || hr >= S) continue;
      const float* xrow = x2 + ((b * 2 + c) * S + hr) * S;
      const float* wrow = sWh + (c * 5 + kk) * S;
      for (int kw = 0; kw < S; ++kw) acc2 = fmaf(xrow[kw], wrow[kw], acc2);
    }
  }
  hw_pre[b * S + t] = acc2;
}

// ---------------------------------------------------------------------------
// Kernel 3: the two [16,128] @ [128,128]^T linears + bias + sigmoid, done with
// V_WMMA_F32_16X16X4_F32.  One block of 512 threads = 16 waves.
// Waves 0..7  -> ww branch, N-tile n0 = wv*16
// Waves 8..15 -> hw branch, N-tile n0 = (wv-8)*16
// Each wave: D[16x16] = A[16x128] x B[128x16], K stepped by 4.
// Fragment layouts per CDNA5 ISA 7.12.2 (32-bit A 16x4 / B 4x16 / C 16x16).
// Branch selection is wave-uniform -> EXEC stays all-ones around WMMA.
// ---------------------------------------------------------------------------
__global__ __launch_bounds__(512) void k3_linear_wmma(
    const float* __restrict__ ww_pre, const float* __restrict__ hw_pre,
    const float* __restrict__ wwW, const float* __restrict__ wwB,
    const float* __restrict__ hwW, const float* __restrict__ hwB,
    float* __restrict__ ww_sig, float* __restrict__ hw_sig) {
  int tid  = threadIdx.x;
  int lane = tid & 31;
  int wv   = tid >> 5;
  bool is_hw = (wv >= 8);

  const float* A  = is_hw ? hw_pre : ww_pre;   // [16,128] row-major (m,k)
  const float* W  = is_hw ? hwW    : wwW;      // [128,128] row-major (n,k)
  const float* Bv = is_hw ? hwB    : wwB;      // [128]
  float*       O  = is_hw ? hw_sig : ww_sig;   // [16,128]

  int n0   = (wv & 7) * 16;
  int m    = lane & 15;              // A row / also N index within tile
  int n    = n0 + m;                 // B/D column for this lane
  int koff = (lane < 16) ? 0 : 2;    // upper half-wave carries K+2..K+3

  v8f c = {};                        // accumulator starts at 0
  for (int kk = 0; kk < S; kk += 4) {
    v2f a, bf;
    a.x  = A[m * S + kk + koff];
    a.y  = A[m * S + kk + koff + 1];
    bf.x = W[n * S + kk + koff];     // B[k][n] = W^T -> W[n][k]
    bf.y = W[n * S + kk + koff + 1];
    c = __builtin_amdgcn_wmma_f32_16x16x4_f32(
        /*neg_a=*/false, a, /*neg_b=*/false, bf,
        /*c_mod=*/(short)0, c, /*reuse_a=*/false, /*reuse_b=*/false);
  }

  float bias  = Bv[n];
  int   mbase = (lane < 16) ? 0 : 8;
  #pragma unroll
  for (int j = 0; j < 8; ++j) {
    float v = c[j] + bias;
    O[(mbase + j) * S + n] = 1.0f / (1.0f + __expf(-v));
  }
}

// ---------------------------------------------------------------------------
// Kernel 4: batched outer product out[b,h,w] = hw[b,h] * ww[b,w].
// ---------------------------------------------------------------------------
__global__ __launch_bounds__(256) void k4_outer(
    const float* __restrict__ hw_sig, const float* __restrict__ ww_sig,
    float* __restrict__ out) {
  int idx = blockIdx.x * 256 + threadIdx.x;   // 0 .. 262143
  int b = idx >> 14;
  int h = (idx >> 7) & 127;
  int w = idx & 127;
  out[idx] = hw_sig[b * S + h] * ww_sig[b * S + w];
}

// ---------------------------------------------------------------------------
extern "C" void kernel_launch(void* const* d_in, const int* in_sizes, int n_in,
                              void* d_out, int out_size, void* d_ws, size_t ws_size,
                              hipStream_t stream) {
  const float* x      = (const float*)d_in[0];  // [16,512,128,128]
  const float* w_conv = (const float*)d_in[1];  // [2,512,1,1]
  const float* w_ww   = (const float*)d_in[2];  // [1,2,128,5]
  const float* w_hw   = (const float*)d_in[3];  // [1,2,5,128]
  const float* wwW    = (const float*)d_in[4];  // [128,128]
  const float* wwB    = (const float*)d_in[5];  // [128]
  const float* hwW    = (const float*)d_in[6];  // [128,128]
  const float* hwB    = (const float*)d_in[7];  // [128]
  float* out = (float*)d_out;

  float* ws     = (float*)d_ws;
  float* x2     = ws;                     // 16*2*128*128 = 524288 floats (2 MB)
  float* ww_pre = ws + 524288;            // 2048
  float* hw_pre = ws + 526336;            // 2048
  float* ww_sig = ws + 528384;            // 2048
  float* hw_sig = ws + 530432;            // 2048

  k1_pointwise<<<256, 256, 0, stream>>>(x, w_conv, x2);
  k2_squeeze<<<BATCH, 128, 0, stream>>>(x2, w_ww, w_hw, ww_pre, hw_pre);
  k3_linear_wmma<<<1, 512, 0, stream>>>(ww_pre, hw_pre, wwW, wwB, hwW, hwB,
                                        ww_sig, hw_sig);
  k4_outer<<<(BATCH * S * S) / 256, 256, 0, stream>>>(hw_sig, ww_sig, out);
}